// ArcRuleFamilyGNN_11974368821434
// MI455X (gfx1250) — compile-verified
//
#include <hip/hip_runtime.h>
#include <hip/hip_bf16.h>

typedef __attribute__((ext_vector_type(2))) float v2f;
typedef __attribute__((ext_vector_type(8))) float v8f;

#define IN_DIM 128
#define HID 64
#define OUT_DIM 10
#define N_GRAPHS 64

// ---------------- degree / normalization ----------------

__global__ void init_deg_kernel(float* deg, int n) {
    int i = blockIdx.x * blockDim.x + threadIdx.x;
    if (i < n) deg[i] = 1.0f;  // self-loop contribution
}

__global__ void deg_count_kernel(const long long* __restrict__ dst, float* deg, int nE) {
    int e = blockIdx.x * blockDim.x + threadIdx.x;
    if (e < nE) atomicAdd(&deg[(int)dst[e]], 1.0f);
}

__global__ void dinv_kernel(const float* __restrict__ deg, float* dinv, int n) {
    int i = blockIdx.x * blockDim.x + threadIdx.x;
    if (i < n) {
        float d = deg[i];
        dinv[i] = d > 0.0f ? rsqrtf(d) : 0.0f;
    }
}

// ---------------- WMMA fp32 GEMM: out[n,64] = A[n,K] @ W[K,64] ----------------
// Block = 256 threads = 8 waves; each wave computes one 16-row tile (4 N-tiles of 16).
// W staged in LDS pre-interleaved in K-pairs so each B fragment is one aligned
// ds_load_b64 landing directly in WMMA operand order (no repack movs).
// A streamed from global: each lane walks its row sequentially (full cacheline reuse).

template <int K>
__global__ __launch_bounds__(256) void gemm_wmma_kernel(const float* __restrict__ A,
                                                        const float* __restrict__ W,
                                                        float* __restrict__ out,
                                                        int nrows) {
    // sW2[p*HID + n] = { W[2p][n], W[2p+1][n] }
    __shared__ v2f sW2[(K / 2) * HID];
    for (int j = threadIdx.x; j < (K / 2) * HID; j += 256) {
        int p = j >> 6;          // K-pair index
        int n = j & 63;          // output column
        v2f w;
        w.x = W[(2 * p) * HID + n];
        w.y = W[(2 * p + 1) * HID + n];
        sW2[j] = w;
    }
    __syncthreads();

    const int wave = threadIdx.x >> 5;
    const int lane = threadIdx.x & 31;
    const int m = lane & 15;       // row within tile (A), col within N-tile (B/C/D)
    const int kh = lane >> 4;      // K-half select: K-offset 2*kh (pair offset kh)

    const int tile = blockIdx.x * 8 + wave;
    const int rowBase = tile * 16;
    if (rowBase < nrows) {
        v8f acc0 = {}, acc1 = {}, acc2 = {}, acc3 = {};
        const float* arow = A + (size_t)(rowBase + m) * K + 2 * kh;
        const v2f* wbase = sW2 + kh * HID + m;   // add p0*HID per k-step

        for (int k0 = 0; k0 < K; k0 += 4) {
            // A frag: lane holds A[M][k0+2*kh], A[M][k0+2*kh+1] (contiguous -> b64)
            const v2f a = *(const v2f*)(arow + k0);
            const v2f* wp = wbase + (k0 >> 1) * HID;
            // B frag per N-tile: lane holds W[k0+2*kh][nt*16+m], W[k0+2*kh+1][...]
            v2f b;
            b = wp[0];
            acc0 = __builtin_amdgcn_wmma_f32_16x16x4_f32(false, a, false, b, (short)0, acc0, false, false);
            b = wp[16];
            acc1 = __builtin_amdgcn_wmma_f32_16x16x4_f32(false, a, false, b, (short)0, acc1, false, false);
            b = wp[32];
            acc2 = __builtin_amdgcn_wmma_f32_16x16x4_f32(false, a, false, b, (short)0, acc2, false, false);
            b = wp[48];
            acc3 = __builtin_amdgcn_wmma_f32_16x16x4_f32(false, a, false, b, (short)0, acc3, false, false);
        }

        // C/D layout: VGPR r -> row M = r + 8*kh ; col N = nt*16 + m
        #pragma unroll
        for (int r = 0; r < 8; ++r) {
            float* o = out + (size_t)(rowBase + r + 8 * kh) * HID + m;
            o[0]  = acc0[r];
            o[16] = acc1[r];
            o[32] = acc2[r];
            o[48] = acc3[r];
        }
    }
}

// ---------------- edge scatter: agg[dst] += xw[src] * dinv[src]*dinv[dst] ----------------
// One thread per (edge, float4 chunk): 16 chunks x 4 floats = 64 channels.
// xw/agg (12.8 MB each) are L2-resident on MI455X (192 MB L2): gathers + fp32
// atomics stay on-chip.

__global__ void scatter_kernel(const long long* __restrict__ src,
                               const long long* __restrict__ dst,
                               const float* __restrict__ dinv,
                               const float* __restrict__ xw,
                               float* __restrict__ agg, int nE) {
    int t = blockIdx.x * blockDim.x + threadIdx.x;
    int e = t >> 4;
    if (e >= nE) return;
    int c = (t & 15) * 4;
    int s = (int)src[e];
    int d = (int)dst[e];
    float w = dinv[s] * dinv[d];
    const float4 v = *(const float4*)(xw + (size_t)s * HID + c);
    float* o = agg + (size_t)d * HID + c;
    atomicAdd(o + 0, v.x * w);
    atomicAdd(o + 1, v.y * w);
    atomicAdd(o + 2, v.z * w);
    atomicAdd(o + 3, v.w * w);
}

// ---------------- finalize: h = relu(agg + xw*dinv^2 + bias) ----------------

__global__ void finalize_kernel(const float* __restrict__ agg,
                                const float* __restrict__ xw,
                                const float* __restrict__ dinv,
                                const float* __restrict__ bias,
                                float* __restrict__ h, int n) {
    int i = blockIdx.x * blockDim.x + threadIdx.x;
    if (i >= n * HID) return;
    int node = i >> 6;
    int c = i & 63;
    float di = dinv[node];
    float v = agg[i] + xw[i] * di * di + bias[c];
    h[i] = fmaxf(v, 0.0f);
}

// ---------------- mean pool over graphs ----------------

__global__ void pool_kernel(const float* __restrict__ h,
                            const long long* __restrict__ batch,
                            float* __restrict__ pooled, float* __restrict__ cnt, int n) {
    int i = blockIdx.x * blockDim.x + threadIdx.x;
    if (i >= n * HID) return;
    int node = i >> 6;
    int c = i & 63;
    int g = (int)batch[node];
    atomicAdd(&pooled[g * HID + c], h[i]);
    if (c == 0) atomicAdd(&cnt[g], 1.0f);
}

__global__ void logits_kernel(const float* __restrict__ pooled,
                              const float* __restrict__ cnt,
                              const float* __restrict__ Wl,
                              const float* __restrict__ bl,
                              float* __restrict__ out) {
    int i = blockIdx.x * blockDim.x + threadIdx.x;
    if (i >= N_GRAPHS * OUT_DIM) return;
    int g = i / OUT_DIM;
    int o = i % OUT_DIM;
    float invc = 1.0f / fmaxf(cnt[g], 1.0f);
    float acc = bl[o];
    for (int c = 0; c < HID; ++c)
        acc += pooled[g * HID + c] * invc * Wl[c * OUT_DIM + o];
    out[i] = acc;
}

// ---------------- launch ----------------

static inline size_t align256(size_t x) { return (x + 255) & ~(size_t)255; }

extern "C" void kernel_launch(void* const* d_in, const int* in_sizes, int n_in,
                              void* d_out, int out_size, void* d_ws, size_t ws_size,
                              hipStream_t stream) {
    const float* x  = (const float*)d_in[0];
    const float* W1 = (const float*)d_in[1];
    const float* b1 = (const float*)d_in[2];
    const float* W2 = (const float*)d_in[3];
    const float* b2 = (const float*)d_in[4];
    const float* Wl = (const float*)d_in[5];
    const float* bl = (const float*)d_in[6];
    const long long* edge_index = (const long long*)d_in[7];  // int64, [2,E]
    const long long* batch      = (const long long*)d_in[8];  // int64, [N]

    const int N  = in_sizes[0] / IN_DIM;
    const int E  = in_sizes[7] / 2;
    const long long* src = edge_index;
    const long long* dst = edge_index + E;

    // workspace layout
    char* ws = (char*)d_ws;
    size_t off = 0;
    float* xw   = (float*)(ws + off); off += align256((size_t)N * HID * sizeof(float));
    float* agg  = (float*)(ws + off); off += align256((size_t)N * HID * sizeof(float));
    float* h    = (float*)(ws + off); off += align256((size_t)N * HID * sizeof(float));
    float* deg  = (float*)(ws + off); off += align256((size_t)N * sizeof(float));
    float* dinv = (float*)(ws + off); off += align256((size_t)N * sizeof(float));
    float* pooled = (float*)(ws + off); off += (size_t)N_GRAPHS * HID * sizeof(float);
    float* cnt    = (float*)(ws + off); off += (size_t)N_GRAPHS * sizeof(float);
    (void)ws_size; (void)n_in; (void)out_size;

    const int T = 256;
    const int nTiles = (N + 15) / 16;
    const int gemmBlocks = (nTiles + 7) / 8;

    // degrees & normalization
    init_deg_kernel<<<(N + T - 1) / T, T, 0, stream>>>(deg, N);
    deg_count_kernel<<<(E + T - 1) / T, T, 0, stream>>>(dst, deg, E);
    dinv_kernel<<<(N + T - 1) / T, T, 0, stream>>>(deg, dinv, N);

    // ---- layer 1 ----
    gemm_wmma_kernel<IN_DIM><<<gemmBlocks, T, 0, stream>>>(x, W1, xw, N);
    hipMemsetAsync(agg, 0, (size_t)N * HID * sizeof(float), stream);
    scatter_kernel<<<((size_t)E * 16 + T - 1) / T, T, 0, stream>>>(src, dst, dinv, xw, agg, E);
    finalize_kernel<<<((size_t)N * HID + T - 1) / T, T, 0, stream>>>(agg, xw, dinv, b1, h, N);

    // ---- layer 2 ----
    gemm_wmma_kernel<HID><<<gemmBlocks, T, 0, stream>>>(h, W2, xw, N);
    hipMemsetAsync(agg, 0, (size_t)N * HID * sizeof(float), stream);
    scatter_kernel<<<((size_t)E * 16 + T - 1) / T, T, 0, stream>>>(src, dst, dinv, xw, agg, E);
    finalize_kernel<<<((size_t)N * HID + T - 1) / T, T, 0, stream>>>(agg, xw, dinv, b2, h, N);

    // ---- pool + head ----
    hipMemsetAsync(pooled, 0, (size_t)(N_GRAPHS * HID + N_GRAPHS) * sizeof(float), stream);
    pool_kernel<<<((size_t)N * HID + T - 1) / T, T, 0, stream>>>(h, batch, pooled, cnt, N);
    logits_kernel<<<(N_GRAPHS * OUT_DIM + T - 1) / T, T, 0, stream>>>(pooled, cnt, Wl, bl, (float*)d_out);
}